// GCNAnomaly_83459804495950
// MI455X (gfx1250) — compile-verified
//
#include <hip/hip_runtime.h>

// Problem constants (match reference setup_inputs)
#define NNODES 100000
#define NEDGES 1600000
#define CIN_   32
#define CH     64
#define CL     16
#define NG     64

typedef float v2f __attribute__((ext_vector_type(2)));
typedef float v8f __attribute__((ext_vector_type(8)));

// Native f32 global atomic add (no return) — guaranteed global_atomic_add_f32,
// never a CAS loop.
__device__ __forceinline__ void atomAddF(float* p, float v) {
    asm volatile("global_atomic_add_f32 %0, %1, off" :: "v"(p), "v"(v) : "memory");
}

// ---------------------------------------------------------------------------
// Elementwise helpers
// ---------------------------------------------------------------------------
__global__ void fill_u32(unsigned* __restrict__ p, unsigned v, int n) {
    int i = blockIdx.x * blockDim.x + threadIdx.x;
    if (i < n) p[i] = v;
}

__global__ void edge_degree(const int* __restrict__ dst, unsigned* __restrict__ degc, int nE) {
    int e = blockIdx.x * blockDim.x + threadIdx.x;
    if (e < nE) atomicAdd(&degc[dst[e]], 1u);   // native global_atomic_add_u32
}

// dinv = rsqrt(count + 1)  (self-loop included)
__global__ void rsqrt_k(const unsigned* __restrict__ degc, float* __restrict__ dinv, int n) {
    int i = blockIdx.x * blockDim.x + threadIdx.x;
    if (i < n) dinv[i] = rsqrtf((float)degc[i] + 1.0f);
}

// out4[i] = act(in4[i] + bias[(4i)%C ..]) ; in==out allowed; total4 = N*C/4
template<int C, bool RELU>
__global__ void bias_act4(const float4* __restrict__ in, const float* __restrict__ bias,
                          float4* __restrict__ out, int total4) {
    int i = blockIdx.x * blockDim.x + threadIdx.x;
    if (i >= total4) return;
    float4 v = in[i];
    const float4 b = *(const float4*)(bias + ((i * 4) & (C - 1)));
    v.x += b.x; v.y += b.y; v.z += b.z; v.w += b.w;
    if (RELU) {
        v.x = fmaxf(v.x, 0.0f); v.y = fmaxf(v.y, 0.0f);
        v.z = fmaxf(v.z, 0.0f); v.w = fmaxf(v.w, 0.0f);
    }
    out[i] = v;
}

// ---------------------------------------------------------------------------
// GEMM: for each node row n:  xw = X[n][:] @ W            (V_WMMA_F32_16X16X4_F32)
// Fused epilogue stores BOTH GCN aggregation operands:
//   XWs[n][c] = xw * dinv[n]          (edge term, pre-scaled by source norm)
//   ACC[n][c] = xw * dinv[n]^2        (self-loop init of the accumulator)
// One wave = one 16x16 tile; 8 waves/block; weights staged in LDS.
// ---------------------------------------------------------------------------
template<int TCIN, int TCOUT>
__global__ __launch_bounds__(256) void gemm_wmma_f32(
    const float* __restrict__ X, const float* __restrict__ W,
    float* __restrict__ XWs, const float* __restrict__ dinv,
    float* __restrict__ ACC, int nrows)
{
    __shared__ float Wlds[TCIN * TCOUT];
    for (int i = threadIdx.x; i < TCIN * TCOUT; i += 256) Wlds[i] = W[i];
    __syncthreads();

    constexpr int COLT = TCOUT / 16;   // column tiles
    constexpr int ROWT = 8 / COLT;     // row tiles per block (8 waves)
    constexpr int RPB  = 16 * ROWT;    // rows covered per block

    const int wave    = threadIdx.x >> 5;   // wave32
    const int lane    = threadIdx.x & 31;
    const int colTile = wave % COLT;
    const int rowTile = wave / COLT;
    const int rowBase = blockIdx.x * RPB + rowTile * 16;

    const int mrow = rowBase + (lane & 15);
    const int srow = mrow < nrows ? mrow : (nrows - 1);  // clamp: keep EXEC all-ones
    const int kq   = (lane >> 4) * 2;                    // K sub-offset per half-wave
    const int ncol = colTile * 16 + (lane & 15);

    const float* xrow = X + (size_t)srow * TCIN;

    v8f acc = {};
#pragma unroll
    for (int kk = 0; kk < TCIN; kk += 4) {
        v2f a = *(const v2f*)(xrow + kk + kq);           // A[m][k], A[m][k+1]
        v2f b;
        b.x = Wlds[(kk + kq)     * TCOUT + ncol];        // W[k][ncol]
        b.y = Wlds[(kk + kq + 1) * TCOUT + ncol];        // W[k+1][ncol]
        acc = __builtin_amdgcn_wmma_f32_16x16x4_f32(
            false, a, false, b, (short)0, acc, false, false);
    }

    // D layout: VGPR r -> row r (lanes 0-15) / r+8 (lanes 16-31), col = lane&15
    const int rAdd = (lane >> 4) * 8;
    if (rowBase + 16 <= nrows) {
        // fast path: whole tile in range (all blocks when N % RPB covers tile)
#pragma unroll
        for (int r = 0; r < 8; ++r) {
            int orow = rowBase + rAdd + r;
            float di = dinv[orow];
            float v  = acc[r] * di;
            XWs[(size_t)orow * TCOUT + ncol] = v;
            ACC[(size_t)orow * TCOUT + ncol] = v * di;
        }
    } else {
#pragma unroll
        for (int r = 0; r < 8; ++r) {
            int orow = rowBase + rAdd + r;
            if (orow < nrows) {
                float di = dinv[orow];
                float v  = acc[r] * di;
                XWs[(size_t)orow * TCOUT + ncol] = v;
                ACC[(size_t)orow * TCOUT + ncol] = v * di;
            }
        }
    }
}

// ---------------------------------------------------------------------------
// Edge scatter: acc[dst][c] += XWs[src][c] * dinv[dst]
// One thread per (edge, 4-channel chunk): float4 gather + 4 native f32 atomics
// sharing one 64-bit address register via immediate offsets.
// ---------------------------------------------------------------------------
template<int C>
__global__ __launch_bounds__(256) void edge_scatter(
    const int* __restrict__ src, const int* __restrict__ dst,
    const float* __restrict__ xws, const float* __restrict__ dinv,
    float* __restrict__ acc, int nE)
{
    const int t = blockIdx.x * blockDim.x + threadIdx.x;
    const int e = t / (C / 4);
    if (e >= nE) return;
    const int c4 = (t % (C / 4)) * 4;
    const int s = src[e], d = dst[e];
    const float w = dinv[d];
    const float4 v = *(const float4*)(xws + (size_t)s * C + c4);
    float* o = acc + (size_t)d * C + c4;
    asm volatile(
        "global_atomic_add_f32 %0, %1, off\n\t"
        "global_atomic_add_f32 %0, %2, off offset:4\n\t"
        "global_atomic_add_f32 %0, %3, off offset:8\n\t"
        "global_atomic_add_f32 %0, %4, off offset:12"
        :: "v"(o), "v"(v.x * w), "v"(v.y * w), "v"(v.z * w), "v"(v.w * w)
        : "memory");
}

// ---------------------------------------------------------------------------
// Global mean pool accumulation, tiny head GEMMs, decoder broadcast gather
// ---------------------------------------------------------------------------
__global__ void pool_sum(const float* __restrict__ h, const int* __restrict__ batch,
                         float* __restrict__ gsum, unsigned* __restrict__ gcnt, int total) {
    int t = blockIdx.x * blockDim.x + threadIdx.x;
    if (t >= total) return;
    int node = t >> 6;          // / CH
    int c    = t & 63;          // % CH
    int g = batch[node];
    atomAddF(&gsum[g * CH + c], h[t]);
    if (c == 0) atomicAdd(&gcnt[g], 1u);
}

// g = gsum/cnt ; z = g@Wp + bp ; Dg = relu(z@Wdp + bdp)
__global__ void head_kernel(const float* __restrict__ gsum, const unsigned* __restrict__ gcnt,
                            const float* __restrict__ Wp,  const float* __restrict__ bp,
                            const float* __restrict__ Wdp, const float* __restrict__ bdp,
                            float* __restrict__ z_out, float* __restrict__ Dg) {
    int g = threadIdx.x;
    if (g >= NG) return;
    float cnt = (float)gcnt[g];
    float inv = 1.0f / fmaxf(cnt, 1.0f);
    float grow[CH];
#pragma unroll 4
    for (int h = 0; h < CH; ++h) grow[h] = gsum[g * CH + h] * inv;
    float zrow[CL];
    for (int l = 0; l < CL; ++l) {
        float s = bp[l];
        for (int h = 0; h < CH; ++h) s += grow[h] * Wp[h * CL + l];
        zrow[l] = s;
        z_out[g * CL + l] = s;
    }
    for (int c = 0; c < CH; ++c) {
        float s = bdp[c];
        for (int l = 0; l < CL; ++l) s += zrow[l] * Wdp[l * CH + c];
        Dg[g * CH + c] = fmaxf(s, 0.0f);
    }
}

// d4[t] = Dg4[batch[node]][chunk]   (float4; total4 = N*CH/4)
__global__ void gather_dec4(const float4* __restrict__ Dg, const int* __restrict__ batch,
                            float4* __restrict__ d, int total4) {
    int t = blockIdx.x * blockDim.x + threadIdx.x;
    if (t >= total4) return;
    int node = t >> 4;                       // / (CH/4)
    d[t] = Dg[batch[node] * (CH / 4) + (t & 15)];
}

// ---------------------------------------------------------------------------
// Launch
// ---------------------------------------------------------------------------
static inline int cdiv(long long a, int b) { return (int)((a + b - 1) / b); }

extern "C" void kernel_launch(void* const* d_in, const int* in_sizes, int n_in,
                              void* d_out, int out_size, void* d_ws, size_t ws_size,
                              hipStream_t stream) {
    (void)in_sizes; (void)n_in; (void)out_size; (void)ws_size;

    const float* x       = (const float*)d_in[0];
    const int*   ei      = (const int*)  d_in[1];
    const int*   batch   = (const int*)  d_in[2];
    const float* W_enc0  = (const float*)d_in[3];  const float* b_enc0 = (const float*)d_in[4];
    const float* W_enc1  = (const float*)d_in[5];  const float* b_enc1 = (const float*)d_in[6];
    const float* W_enc2  = (const float*)d_in[7];  const float* b_enc2 = (const float*)d_in[8];
    const float* W_proj  = (const float*)d_in[9];  const float* b_proj = (const float*)d_in[10];
    const float* W_dproj = (const float*)d_in[11]; const float* b_dproj= (const float*)d_in[12];
    const float* W_dec0  = (const float*)d_in[13]; const float* b_dec0 = (const float*)d_in[14];
    const float* W_dec1  = (const float*)d_in[15]; const float* b_dec1 = (const float*)d_in[16];

    const int* src = ei;            // edge_index[0]
    const int* dst = ei + NEDGES;   // edge_index[1]
    float* out = (float*)d_out;

    // Workspace layout: 3 big N×64 f32 buffers + degree/pool scratch (~77 MB)
    float* w  = (float*)d_ws;
    const size_t NC = (size_t)NNODES * CH;
    float*    B0   = w;                         // xw*dinv staging
    float*    B1   = w + NC;                    // acc / activation ping
    float*    B2   = w + 2 * NC;                // acc / activation pong
    unsigned* degc = (unsigned*)(w + 3 * NC);   // edge counts (int)
    float*    dinv = (float*)(degc + NNODES);
    float*    gsum = dinv + NNODES;             // NG*CH
    unsigned* gcnt = (unsigned*)(gsum + NG * CH);
    float*    Dg   = (float*)(gcnt + NG);       // NG*CH

    const int gb64 = cdiv(NNODES, 32);               // <*,64> gemm blocks
    const int gb32 = cdiv(NNODES, 64);               // <*,32> gemm blocks
    const int ew64 = cdiv((long long)NNODES * CH,   256);
    const int e464 = cdiv((long long)NNODES * CH / 4,   256);
    const int e432 = cdiv((long long)NNODES * CIN_ / 4, 256);
    const int sc64 = cdiv((long long)NEDGES * (CH / 4),   256);
    const int sc32 = cdiv((long long)NEDGES * (CIN_ / 4), 256);

    // --- degrees & normalization ---
    fill_u32<<<cdiv(NNODES, 256), 256, 0, stream>>>(degc, 0u, NNODES);
    fill_u32<<<cdiv(NG * CH + NG, 256), 256, 0, stream>>>((unsigned*)gsum, 0u, NG * CH + NG);
    edge_degree<<<cdiv(NEDGES, 256), 256, 0, stream>>>(dst, degc, NEDGES);
    rsqrt_k<<<cdiv(NNODES, 256), 256, 0, stream>>>(degc, dinv, NNODES);

    // --- encoder conv 0: x(32) -> B1 ---
    gemm_wmma_f32<32, 64><<<gb64, 256, 0, stream>>>(x, W_enc0, B0, dinv, B1, NNODES);
    edge_scatter<64><<<sc64, 256, 0, stream>>>(src, dst, B0, dinv, B1, NEDGES);
    bias_act4<64, true><<<e464, 256, 0, stream>>>((float4*)B1, b_enc0, (float4*)B1, NNODES * CH / 4);

    // --- encoder conv 1: B1 -> B2 ---
    gemm_wmma_f32<64, 64><<<gb64, 256, 0, stream>>>(B1, W_enc1, B0, dinv, B2, NNODES);
    edge_scatter<64><<<sc64, 256, 0, stream>>>(src, dst, B0, dinv, B2, NEDGES);
    bias_act4<64, true><<<e464, 256, 0, stream>>>((float4*)B2, b_enc1, (float4*)B2, NNODES * CH / 4);

    // --- encoder conv 2: B2 -> B1 ---
    gemm_wmma_f32<64, 64><<<gb64, 256, 0, stream>>>(B2, W_enc2, B0, dinv, B1, NNODES);
    edge_scatter<64><<<sc64, 256, 0, stream>>>(src, dst, B0, dinv, B1, NEDGES);
    bias_act4<64, true><<<e464, 256, 0, stream>>>((float4*)B1, b_enc2, (float4*)B1, NNODES * CH / 4);

    // --- pool + latent head + broadcast ---
    pool_sum<<<ew64, 256, 0, stream>>>(B1, batch, gsum, gcnt, NNODES * CH);
    head_kernel<<<1, 64, 0, stream>>>(gsum, gcnt, W_proj, b_proj, W_dproj, b_dproj,
                                      out + (size_t)NNODES * CIN_, Dg);
    gather_dec4<<<e464, 256, 0, stream>>>((const float4*)Dg, batch, (float4*)B2, NNODES * CH / 4);

    // --- decoder conv 0: B2 -> B1 ---
    gemm_wmma_f32<64, 64><<<gb64, 256, 0, stream>>>(B2, W_dec0, B0, dinv, B1, NNODES);
    edge_scatter<64><<<sc64, 256, 0, stream>>>(src, dst, B0, dinv, B1, NEDGES);
    bias_act4<64, true><<<e464, 256, 0, stream>>>((float4*)B1, b_dec0, (float4*)B1, NNODES * CH / 4);

    // --- decoder conv 1: B1 -> x_recon (out), no relu ---
    gemm_wmma_f32<64, 32><<<gb32, 256, 0, stream>>>(B1, W_dec1, B0, dinv, B2, NNODES);
    edge_scatter<32><<<sc32, 256, 0, stream>>>(src, dst, B0, dinv, B2, NEDGES);
    bias_act4<32, false><<<e432, 256, 0, stream>>>((float4*)B2, b_dec1, (float4*)out, NNODES * CIN_ / 4);
}